// local_scan_zero_ones_81303730913732
// MI455X (gfx1250) — compile-verified
//
#include <hip/hip_runtime.h>
#include <cstdint>

// Problem constants (B,C,H,W,K = 4,64,512,512,8)
#define B_    4
#define C_    64
#define H_    512
#define W_    512
#define K_    8
#define NWIN_ 4096                 // (H/K)*(W/K) windows per batch
#define M_    ((C_ * NWIN_) / 2)   // 131072 compacted columns per half

typedef float v4f __attribute__((ext_vector_type(4)));

// ---------------------------------------------------------------------------
// Kernel 1: per-window "any(mask>0)" flag.  4 MB read, trivial.
// ---------------------------------------------------------------------------
__global__ __launch_bounds__(256) void win_flags_kernel(const float* __restrict__ mask,
                                                        int* __restrict__ flags) {
  int gw = blockIdx.x * 256 + threadIdx.x;   // [0, B*NWIN)
  int b  = gw >> 12;                          // / 4096
  int n  = gw & (NWIN_ - 1);
  int ph = n >> 6, pw = n & 63;
  const float* p = mask + ((size_t)b * H_ + (size_t)ph * K_) * W_ + (size_t)pw * K_;
  int any = 0;
#pragma unroll
  for (int kh = 0; kh < K_; ++kh) {
    const v4f* row = (const v4f*)(p + (size_t)kh * W_);
    v4f a = row[0], c = row[1];
    any |= (a.x > 0.f) | (a.y > 0.f) | (a.z > 0.f) | (a.w > 0.f) |
           (c.x > 0.f) | (c.y > 0.f) | (c.z > 0.f) | (c.w > 0.f);
  }
  flags[gw] = any;
}

// ---------------------------------------------------------------------------
// Kernel 2: per-batch exclusive prefix sum of 4096 flags (1 block / batch).
// ---------------------------------------------------------------------------
__global__ __launch_bounds__(1024) void win_scan_kernel(const int* __restrict__ flags,
                                                        int* __restrict__ P,
                                                        int* __restrict__ S) {
  __shared__ int sdata[1024];
  int b   = blockIdx.x;
  int tid = threadIdx.x;
  int idx = b * NWIN_ + tid * 4;
  int v0 = flags[idx], v1 = flags[idx + 1], v2 = flags[idx + 2], v3 = flags[idx + 3];
  int s  = v0 + v1 + v2 + v3;
  sdata[tid] = s;
  __syncthreads();
  for (int off = 1; off < 1024; off <<= 1) {
    int t = (tid >= off) ? sdata[tid - off] : 0;
    __syncthreads();
    sdata[tid] += t;
    __syncthreads();
  }
  int base = sdata[tid] - s;                 // exclusive prefix at idx
  P[idx]     = base;
  P[idx + 1] = base + v0;
  P[idx + 2] = base + v0 + v1;
  P[idx + 3] = base + v0 + v1 + v2;
  if (tid == 1023) S[b] = sdata[1023];
}

// ---------------------------------------------------------------------------
// Kernel 3: the permutation.  Block = (b, c, strip of 4 window-rows).
//   - 64 KB x-strip -> LDS via async global->LDS b128 loads
//   - compaction lists straight from P (block covers a contiguous n-range)
//   - gather from swizzled LDS, stream float4 non-temporal stores
// ---------------------------------------------------------------------------
__device__ __forceinline__ int swz(int h, int w) {
  // row stride 540, +4 every 64 floats: breaks the stride-64 bank pattern,
  // keeps 16B alignment for b128 LDS writes (all terms multiples of 4).
  return h * 540 + w + ((w >> 6) << 2);
}

__global__ __launch_bounds__(256) void permute_kernel(const float* __restrict__ x,
                                                      const int* __restrict__ flags,
                                                      const int* __restrict__ P,
                                                      const int* __restrict__ Stot,
                                                      float* __restrict__ out) {
  __shared__ float smem[32 * 540];           // 32 H-rows x 512 W (padded) = 67.5 KB
  __shared__ unsigned short olist[256];
  __shared__ unsigned short zlist[256];

  int bid = blockIdx.x;
  int pb  = bid & 15;                        // strip index: ph in [4*pb, 4*pb+4)
  int c   = (bid >> 4) & 63;
  int b   = bid >> 10;
  int tid = threadIdx.x;

  const float* src = x + (((size_t)(b * C_ + c)) * H_ + (size_t)pb * 32) * W_;

  // ---- async global -> LDS: 4096 float4 chunks, 16 per thread, coalesced
#pragma unroll
  for (int it = 0; it < 16; ++it) {
    int cidx = tid + 256 * it;
    int h = cidx >> 7;                        // 128 chunks per row
    int w = (cidx & 127) << 2;
    uint32_t lofs = (uint32_t)(uintptr_t)(&smem[swz(h, w)]);   // addr[31:0] = LDS offset
    uint64_t g    = (uint64_t)(uintptr_t)(src + (size_t)h * W_ + w);
    asm volatile("global_load_async_to_lds_b128 %0, %1, off"
                 :: "v"(lofs), "v"(g) : "memory");
  }

  // ---- build stable compaction lists directly from prefix sums
  int n0  = pb * 256;                         // first window of this strip
  int Pn0 = P[b * NWIN_ + n0];
  int S   = Stot[b];
  {
    int n  = n0 + tid;                        // one thread per local window
    int f  = flags[b * NWIN_ + n];
    int pv = P[b * NWIN_ + n];
    if (f) olist[pv - Pn0]               = (unsigned short)tid;
    else   zlist[(n - pv) - (n0 - Pn0)]  = (unsigned short)tid;
  }
  int Pend = (pb == 15) ? S : P[b * NWIN_ + n0 + 256];
  int cntO = Pend - Pn0;

#if __has_builtin(__builtin_amdgcn_s_wait_asynccnt)
  __builtin_amdgcn_s_wait_asynccnt(0);
#else
  asm volatile("s_wait_asynccnt 0" ::: "memory");
#endif
  __syncthreads();

  // ---- scatter: 128 row-tasks (64 r x {ones,zeros}), 16 per wave
  int wave = tid >> 5, lane = tid & 31;
  for (int j = 0; j < 16; ++j) {
    int t  = wave * 16 + j;
    int zs = t & 1;                           // 0 = ones half, 1 = zeros half
    int r  = t >> 1;                          // output row (kh*8+kw)
    int kh = r >> 3, kw = r & 7;

    int cnt; size_t q0; const unsigned short* list; float* base;
    if (!zs) {
      cnt  = cntO;
      q0   = (size_t)c * S + (size_t)Pn0;
      list = olist;
      base = out + ((size_t)(b * 64 + r)) * M_ + q0;
    } else {
      cnt  = 256 - cntO;
      q0   = (size_t)c * (NWIN_ - S) + (size_t)(n0 - Pn0);
      list = zlist;
      base = out + (size_t)B_ * 64 * M_ + ((size_t)(b * 64 + r)) * M_ + q0;
    }

    bool vec = ((q0 & 3) == 0) && ((cnt & 3) == 0);
    for (int i = lane * 4; i < cnt; i += 128) {
      if (vec) {
        int e0 = list[i], e1 = list[i + 1], e2 = list[i + 2], e3 = list[i + 3];
        v4f v;
        v.x = smem[swz((e0 >> 6) * 8 + kh, (e0 & 63) * 8 + kw)];
        v.y = smem[swz((e1 >> 6) * 8 + kh, (e1 & 63) * 8 + kw)];
        v.z = smem[swz((e2 >> 6) * 8 + kh, (e2 & 63) * 8 + kw)];
        v.w = smem[swz((e3 >> 6) * 8 + kh, (e3 & 63) * 8 + kw)];
        __builtin_nontemporal_store(v, (v4f*)(base + i));       // 512B/wave stream
      } else {
#pragma unroll
        for (int u = 0; u < 4; ++u) {
          if (i + u < cnt) {
            int e = list[i + u];
            __builtin_nontemporal_store(
                smem[swz((e >> 6) * 8 + kh, (e & 63) * 8 + kw)], base + i + u);
          }
        }
      }
    }
  }
}

// ---------------------------------------------------------------------------
extern "C" void kernel_launch(void* const* d_in, const int* in_sizes, int n_in,
                              void* d_out, int out_size, void* d_ws, size_t ws_size,
                              hipStream_t stream) {
  (void)in_sizes; (void)n_in; (void)out_size; (void)ws_size;
  const float* mask = (const float*)d_in[0];
  const float* x    = (const float*)d_in[1];
  float*       out  = (float*)d_out;

  int* flags = (int*)d_ws;                                            // 64 KB
  int* P     = (int*)((char*)d_ws + (size_t)B_ * NWIN_ * 4);          // 64 KB
  int* S     = (int*)((char*)d_ws + (size_t)2 * B_ * NWIN_ * 4);      // 16 B

  win_flags_kernel<<<(B_ * NWIN_) / 256, 256, 0, stream>>>(mask, flags);
  win_scan_kernel<<<B_, 1024, 0, stream>>>(flags, P, S);
  permute_kernel<<<B_ * C_ * 16, 256, 0, stream>>>(x, flags, P, S, out);
}